// Spatialflatten_65575560675925
// MI455X (gfx1250) — compile-verified
//
#include <hip/hip_runtime.h>
#include <stdint.h>

// Spatialflatten: fm [B=8, C=256, H=128, W=128] f32 (NCHW)
//   -> out [B, H*W, 9*C] f32, out[b, h*W+w, k*C+c] = fm[b, c, clamp(h+dy-1), clamp(w+dx-1)]
// Pure data movement: 1.21 GB stores + ~134 MB loads (L2-resident) => store-BW bound (~55us floor).
// Strategy: LDS transpose tile. Stage 3 clamped rows x CT channels x (WT+2) pixels into LDS with
// gfx1250 async-to-LDS gathers (ASYNCcnt), then emit lanes-along-channel coalesced NT stores.

#define BB 8
#define CC 256
#define HH 128
#define WW 128
#define K9 9
#define HWX (HH * WW)          // 16384
#define OC (K9 * CC)           // 2304
#define CT 128                 // channel tile
#define WT 32                  // pixel tile (one row segment)
#define JW (WT + 2)            // 34 staged pixels (halo)
#define LDS_ELEMS (3 * CT * JW)        // 13056 floats = 52224 B
#define STAGE_ITERS (LDS_ELEMS / 256)  // 51 (exact, no tail)

#if defined(__gfx1250__)
#define USE_ASYNC_LDS 1
#else
#define USE_ASYNC_LDS 0
#endif

__global__ __launch_bounds__(256) void spatialflatten_kernel(
    const float* __restrict__ fm, float* __restrict__ out)
{
    __shared__ float sh[LDS_ELEMS];    // layout [r=3][c=CT][j=JW], linear

    const int tid = threadIdx.x;
    const int b   = blockIdx.z;                 // 0..7
    const int h   = blockIdx.y;                 // 0..127
    const int w0  = (blockIdx.x & 3) * WT;      // 4 w-tiles
    const int c0  = (blockIdx.x >> 2) * CT;     // 2 c-tiles

    // ---------------- stage: 3 clamped rows, coalesced along w ----------------
    // linear idx = (r*CT + c)*JW + j, walked incrementally (256 = 7*34 + 18)
    int j  = tid % JW;
    int rc = tid / JW;
    for (int i = 0; i < STAGE_ITERS; ++i) {
        const int c = rc & (CT - 1);
        const int r = rc >> 7;
        int wsrc = w0 - 1 + j;
        wsrc = wsrc < 0 ? 0 : (wsrc > WW - 1 ? WW - 1 : wsrc);
        int hsrc = h - 1 + r;
        hsrc = hsrc < 0 ? 0 : (hsrc > HH - 1 ? HH - 1 : hsrc);
        const float* gp = fm + (((size_t)(b * CC + c0 + c)) * HH + hsrc) * WW + wsrc;
        const int idx = rc * JW + j;            // == i*256 + tid
#if USE_ASYNC_LDS
        // gfx1250 async gather: copies the float at global address gp to the LDS address lds_addr, tracked by ASYNCcnt.
        // Flat LDS aperture keeps the workgroup-relative offset in addr[31:0].
        const uint32_t lds_addr = (uint32_t)(uintptr_t)(&sh[idx]);
        asm volatile("global_load_async_to_lds_b32 %0, %1, off"
                     :: "v"(lds_addr), "v"(gp) : "memory");
#else
        sh[idx] = *gp;
#endif
        j += 18;
        rc += 7;
        if (j >= JW) { j -= JW; rc += 1; }
    }
#if USE_ASYNC_LDS
    asm volatile("s_wait_asynccnt 0x0" ::: "memory");
#endif
    __syncthreads();

    // ---------------- emit: lanes along c => 512B contiguous NT stores --------
    // tid = rest0*128 + c_local ; thread covers w = {rest0, rest0+2, ...} x k = 0..8
    const int c_local = tid & (CT - 1);
    const int rest0   = tid >> 7;               // 0 or 1
    const float* lb = &sh[c_local * JW + rest0];
    float* op = out
        + ((size_t)b * HWX + (size_t)h * WW + (size_t)(w0 + rest0)) * OC
        + (size_t)(c0 + c_local);

#pragma unroll
    for (int k = 0; k < K9; ++k) {
        const int dy = k / 3, dx = k % 3;       // compile-time per unrolled k
        const int lk = dy * CT * JW + dx;       // LDS immediate part
        const int ok = k * CC;                  // out immediate part
#pragma unroll
        for (int wi = 0; wi < WT / 2; ++wi) {
            const float v = lb[lk + 2 * wi];    // ds_load_b32, conflict-free (stride 34)
            __builtin_nontemporal_store(v, op + (size_t)(2 * wi) * OC + ok);
        }
    }
}

extern "C" void kernel_launch(void* const* d_in, const int* in_sizes, int n_in,
                              void* d_out, int out_size, void* d_ws, size_t ws_size,
                              hipStream_t stream) {
    (void)in_sizes; (void)n_in; (void)d_ws; (void)ws_size; (void)out_size;
    const float* fm = (const float*)d_in[0];
    float* out = (float*)d_out;
    // grid: x = 4 w-tiles * 2 c-tiles, y = H rows, z = B
    dim3 grid((WW / WT) * (CC / CT), HH, BB);   // (8, 128, 8) = 8192 blocks
    spatialflatten_kernel<<<grid, 256, 0, stream>>>(fm, out);
}